// EncoderSDP_39582418600311
// MI455X (gfx1250) — compile-verified
//
#include <hip/hip_runtime.h>

namespace {
constexpr int kB  = 16;
constexpr int kL  = 512;
constexpr int kP  = 32;
constexpr int kD  = 256;
constexpr int kDC = 64;                // D-slice staged in LDS per workgroup
constexpr int kNDC = kD / kDC;         // 4
constexpr int kLT = 128;               // tokens per workgroup
constexpr int kNLT = kL / kLT;         // 4
constexpr float kNeg = -1e30f;
constexpr int kLdsBytes = kL * kDC * (int)sizeof(float);   // 128 KiB
}

typedef __attribute__((ext_vector_type(4))) unsigned int u32x4_;
typedef __attribute__((ext_vector_type(8))) int          i32x8_;
typedef __attribute__((ext_vector_type(4))) int          i32x4_;

__global__ __launch_bounds__(256) void encoder_sdp_kernel(
    const float* __restrict__ inputs,
    const int*   __restrict__ left_paths,
    const int*   __restrict__ right_paths,
    const int*   __restrict__ left_lens,
    const int*   __restrict__ right_lens,
    const int*   __restrict__ sent_lens,
    float*       __restrict__ out)
{
    extern __shared__ float lds[];

    const int ltile  = blockIdx.x;   // 0..3
    const int dchunk = blockIdx.y;   // 0..3
    const int b      = blockIdx.z;   // 0..15

    // ---- Stage inputs[b, 0:512, dchunk*64 : dchunk*64+64] into LDS with the
    // Tensor Data Mover. 2D tile: tile_dim0=64 elems, tile_dim1=512 rows,
    // row stride = 256 elems, data_size = 4 bytes. Issued by wave 0 only.
    if (threadIdx.x < 32) {
        const unsigned long long gaddr =
            (unsigned long long)(const void*)(inputs +
                ((size_t)b * kL * kD + (size_t)dchunk * kDC));
        const unsigned lds_off = (unsigned)(size_t)(void*)lds;  // AS(3) offset in low 32b

        u32x4_ g0;
        g0[0] = 1u;                                   // count=1, user D#, no gather
        g0[1] = lds_off;                              // lds_addr (bytes)
        g0[2] = (unsigned)(gaddr & 0xffffffffull);    // global_addr[31:0]
        g0[3] = (unsigned)((gaddr >> 32) & 0x1ffffffull) | (2u << 30); // [56:32], type=2

        i32x8_ g1;
        g1[0] = (int)(2u << 16);                      // workgroup_mask=0, data_size=4B
        g1[1] = (int)((unsigned)kD  << 16);           // tensor_dim0 = 256 (lo16)
        g1[2] = (int)((unsigned)kL  << 16);           // dim0 hi=0 | tensor_dim1 = 512 (lo16)
        g1[3] = (int)((unsigned)kDC << 16);           // dim1 hi=0 | tile_dim0 = 64
        g1[4] = kL;                                   // tile_dim1 = 512, tile_dim2 = 0
        g1[5] = kD;                                   // tensor_dim0_stride = 256
        g1[6] = 0;                                    // stride hi, tensor_dim1_stride lo
        g1[7] = 0;

        i32x4_ gz = {0, 0, 0, 0};
#if __clang_major__ >= 23
        i32x8_ gz8 = {0, 0, 0, 0, 0, 0, 0, 0};
        __builtin_amdgcn_tensor_load_to_lds(g0, g1, gz, gz, gz8, 0);
#else
        __builtin_amdgcn_tensor_load_to_lds(g0, g1, gz, gz, 0);
#endif
        __builtin_amdgcn_s_wait_tensorcnt(0);
    }
    __syncthreads();

    // ---- One wave per token; each lane owns 2 consecutive floats of the
    // 64-float D-slice -> ds_load_b64 gathers hit all 64 banks once per wave.
    const int lane = threadIdx.x & 31;
    const int wave = threadIdx.x >> 5;        // 0..7
    const int d0   = lane * 2;

    const int    slen    = sent_lens[b];
    const int    l0      = ltile * kLT;
    const size_t bl_base = (size_t)b * kL;

    for (int li = wave; li < kLT; li += 8) {
        // token index is wave-uniform; make it explicit so index/length reads
        // become scalar loads (KM path) instead of per-lane vector loads.
        const int l = __builtin_amdgcn_readfirstlane(l0 + li);

        float lx0, lx1, rx0, rx1;
        if (l < slen) {
            const size_t tok = bl_base + (size_t)l;
            const int* __restrict__ lp = left_paths  + tok * kP;
            const int* __restrict__ rp = right_paths + tok * kP;
            const int ll = left_lens[tok];
            const int rl = right_lens[tok];

            lx0 = lx1 = kNeg;
#pragma unroll 4
            for (int p = 0; p < ll; ++p) {
                const int idx = lp[p];
                const float2 v = *(const float2*)(lds + (size_t)idx * kDC + d0);
                lx0 = fmaxf(lx0, v.x);
                lx1 = fmaxf(lx1, v.y);
            }
            rx0 = rx1 = kNeg;
#pragma unroll 4
            for (int p = 0; p < rl; ++p) {
                const int idx = rp[p];
                const float2 v = *(const float2*)(lds + (size_t)idx * kDC + d0);
                rx0 = fmaxf(rx0, v.x);
                rx1 = fmaxf(rx1, v.y);
            }
        } else {
            lx0 = lx1 = rx0 = rx1 = 0.0f;     // zero-pad past sentence length
        }

        float* o = out + (bl_base + (size_t)l) * (2 * kD) + dchunk * kDC + d0;
        *(float2*)(o)       = make_float2(lx0, lx1);  // left half
        *(float2*)(o + kD)  = make_float2(rx0, rx1);  // right half
    }
}

extern "C" void kernel_launch(void* const* d_in, const int* in_sizes, int n_in,
                              void* d_out, int out_size, void* d_ws, size_t ws_size,
                              hipStream_t stream) {
    const float* inputs      = (const float*)d_in[0];
    const int*   left_paths  = (const int*)d_in[1];
    const int*   right_paths = (const int*)d_in[2];
    const int*   left_lens   = (const int*)d_in[3];
    const int*   right_lens  = (const int*)d_in[4];
    const int*   sent_lens   = (const int*)d_in[5];
    float*       out         = (float*)d_out;

    dim3 grid(kNLT, kNDC, kB);   // 4 x 4 x 16 = 256 workgroups
    encoder_sdp_kernel<<<grid, 256, kLdsBytes, stream>>>(
        inputs, left_paths, right_paths, left_lens, right_lens, sent_lens, out);
}